// basicGNN_89790586290566
// MI455X (gfx1250) — compile-verified
//
#include <hip/hip_runtime.h>
#include <stdint.h>

#define N_NODES 50000
#define N_EDGES 800000
#define D 64
#define ROW_BLKS (N_NODES / 16)   // 3125, exact
#define TOTAL_WAVES (ROW_BLKS * 4)

typedef __attribute__((ext_vector_type(2))) float v2f;
typedef __attribute__((ext_vector_type(8))) float v8f;

// ---------------------------------------------------------------------------
// Scatter-add: agg[dst[e]] += x[src[e]]  (64 f32 per edge, 16 lanes/edge,
// float4 gather + 4x global_atomic_add_f32; agg stays L2-resident)
// ---------------------------------------------------------------------------
__global__ __launch_bounds__(256) void gnn_scatter_add(
    const float* __restrict__ x,
    const long long* __restrict__ src,
    const long long* __restrict__ dst,
    float* __restrict__ agg)
{
    int tid = blockIdx.x * 256 + threadIdx.x;
    int e   = tid >> 4;
    int c   = (tid & 15) * 4;
    if (e >= N_EDGES) return;
    long long s = src[e];
    long long d = dst[e];
    const float4 v = *(const float4*)(x + s * D + c);
    float* p = agg + d * D + c;
    atomicAdd(p + 0, v.x);
    atomicAdd(p + 1, v.y);
    atomicAdd(p + 2, v.z);
    atomicAdd(p + 3, v.w);
}

// ---------------------------------------------------------------------------
// Fused GraphConv transform:
//   out = relu?( agg @ W_rel^T + b_rel + x @ W_root^T )
// One wave -> 16-node x 16-feature tile; both GEMMs chained into one f32
// accumulator via V_WMMA_F32_16X16X4_F32 (16 k-steps x 2 WMMAs).
// ---------------------------------------------------------------------------
__global__ __launch_bounds__(256) void gnn_wmma_transform(
    const float* __restrict__ agg,
    const float* __restrict__ x,
    const float* __restrict__ w_rel,
    const float* __restrict__ b_rel,
    const float* __restrict__ w_root,
    float* __restrict__ out,
    int do_relu)
{
    const int lane = threadIdx.x & 31;
    const int wid  = threadIdx.x >> 5;
    const int gw   = blockIdx.x * 8 + wid;
    if (gw >= TOTAL_WAVES) return;   // wave-uniform: EXEC all-1s inside

    const int row_blk = gw >> 2;       // which 16-node block
    const int ntile   = gw & 3;        // which 16-feature block
    const int half    = lane >> 4;     // 0: K=0,1 | 1: K=2,3
    const int l       = lane & 15;

    // --- preload B operands (weights) into registers ------------------------
    // B[k][n] = W[n][k]; lane l holds column n = ntile*16 + l.
    v2f brel[16], broot[16];
    const int wrow = ntile * 16 + l;
    #pragma unroll
    for (int kk = 0; kk < 16; ++kk) {
        const int off = kk * 4 + half * 2;
        brel[kk]  = *(const v2f*)(w_rel  + wrow * D + off);
        broot[kk] = *(const v2f*)(w_root + wrow * D + off);
    }
    const float bias = b_rel[ntile * 16 + l];

    // --- K loop: 2 WMMAs per k-step, single chained accumulator -------------
    v8f acc = {};
    const int arow = row_blk * 16 + l;   // A row M = lane&15
    #pragma unroll
    for (int kk = 0; kk < 16; ++kk) {
        const int off = kk * 4 + half * 2;
        v2f a_agg = *(const v2f*)(agg + (long long)arow * D + off);
        v2f a_x   = *(const v2f*)(x   + (long long)arow * D + off);
        acc = __builtin_amdgcn_wmma_f32_16x16x4_f32(
                  false, a_agg, false, brel[kk],  (short)0, acc, false, false);
        acc = __builtin_amdgcn_wmma_f32_16x16x4_f32(
                  false, a_x,   false, broot[kk], (short)0, acc, false, false);
    }

    // --- epilogue: bias (+ReLU) + store -------------------------------------
    // C/D layout: VGPR r, lanes 0-15 -> M=r ; lanes 16-31 -> M=r+8 ; N=l.
    #pragma unroll
    for (int r = 0; r < 8; ++r) {
        float v = acc[r] + bias;
        if (do_relu) v = fmaxf(v, 0.0f);
        out[(long long)(row_blk * 16 + half * 8 + r) * D + ntile * 16 + l] = v;
    }
}

// ---------------------------------------------------------------------------
extern "C" void kernel_launch(void* const* d_in, const int* in_sizes, int n_in,
                              void* d_out, int out_size, void* d_ws, size_t ws_size,
                              hipStream_t stream) {
    const float*     x   = (const float*)d_in[0];
    const long long* ei  = (const long long*)d_in[1];   // int64 (2, N_EDGES)
    const long long* src = ei;
    const long long* dst = ei + N_EDGES;

    const float* w_rel[3]  = { (const float*)d_in[2], (const float*)d_in[5], (const float*)d_in[8] };
    const float* b_rel[3]  = { (const float*)d_in[3], (const float*)d_in[6], (const float*)d_in[9] };
    const float* w_root[3] = { (const float*)d_in[4], (const float*)d_in[7], (const float*)d_in[10] };

    const size_t NF = (size_t)N_NODES * D;
    float* agg = (float*)d_ws;
    float* h0  = agg + NF;
    float* h1  = h0 + NF;
    float* out = (float*)d_out;

    const size_t aggBytes = NF * sizeof(float);

    const dim3 sblk(256);
    const dim3 sgrd((N_EDGES * 16 + 255) / 256);
    const dim3 gblk(256);
    const dim3 ggrd((TOTAL_WAVES + 7) / 8);

    const float* ins[3]  = { x,  h0, h1 };
    float*       outs[3] = { h0, h1, out };

    for (int layer = 0; layer < 3; ++layer) {
        hipMemsetAsync(agg, 0, aggBytes, stream);
        gnn_scatter_add<<<sgrd, sblk, 0, stream>>>(ins[layer], src, dst, agg);
        gnn_wmma_transform<<<ggrd, gblk, 0, stream>>>(
            agg, ins[layer], w_rel[layer], b_rel[layer], w_root[layer],
            outs[layer], (layer < 2) ? 1 : 0);
    }
}